// AttentionDist_87789131530406
// MI455X (gfx1250) — compile-verified
//
#include <hip/hip_runtime.h>

typedef __attribute__((ext_vector_type(2))) float v2f;
typedef __attribute__((ext_vector_type(8))) float v8f;

#define WAVES 8
#define TPW   4
#define ROWS_PER_BLOCK (WAVES * TPW * 16) /* 512 */
#define NEGBIG (-9000000000000000.0f)

__device__ __forceinline__ void online_combine(float& m, float& s, float m2, float s2) {
    float M = fmaxf(m, m2);
    s = s * __expf(m - M) + s2 * __expf(m2 - M);
    m = M;
}

// Kernel 1: e[b,n] = leaky_relu(tgt[b] + x[b,n,:]·a_src) via v_wmma_f32_16x16x4_f32,
// plus per-(b,chunk) online-softmax partials (max, sum).
__global__ void gat_scores_kernel(const float* __restrict__ x,
                                  const float* __restrict__ a,
                                  const int*   __restrict__ node_index_p,
                                  const int*   __restrict__ adj_mask,
                                  float*       __restrict__ e_out,
                                  float2*      __restrict__ partials,
                                  int N, int nchunks)
{
    const int b     = blockIdx.y;
    const int chunk = blockIdx.x;
    const int tid   = threadIdx.x;
    const int lane  = tid & 31;
    const int wave  = tid >> 5;
    const int hi    = lane >> 4;   // 0: lanes 0-15, 1: lanes 16-31
    const int lo    = lane & 15;

    const int node_index = node_index_p[0];
    const size_t xb = (size_t)b * (size_t)N * 64;

    // tgt_score[b] = x[b,node_index,:] . a_tgt  (32 lanes x 2 elements, shfl reduce)
    float tgt;
    {
        const float* xr = x + xb + (size_t)node_index * 64 + 2 * lane;
        float p = xr[0] * a[2 * lane] + xr[1] * a[2 * lane + 1];
        #pragma unroll
        for (int off = 16; off > 0; off >>= 1) p += __shfl_xor(p, off, 32);
        tgt = p;
    }

    // Preload a_src in WMMA A-operand layout: asv[k][v] = a_src[4k + 2*hi + v]
    v2f asv[16];
    #pragma unroll
    for (int k = 0; k < 16; ++k) {
        const float* ap = a + 64 + 4 * k + 2 * hi;
        v2f t; t[0] = ap[0]; t[1] = ap[1];
        asv[k] = t;
    }

    float vals[TPW];
    #pragma unroll
    for (int i = 0; i < TPW; ++i) {
        vals[i] = NEGBIG;
        const int tile = chunk * (WAVES * TPW) + i * WAVES + wave;
        const int n0   = tile * 16;
        if (n0 < N) {  // wave-uniform: N % 16 == 0, so full tile valid; EXEC all-1s at WMMA
            const float* xrow = x + xb + (size_t)(n0 + lo) * 64 + 2 * hi;
            v8f c = {};
            #pragma unroll
            for (int k = 0; k < 16; ++k) {
                v2f xv = *(const v2f*)(xrow + 4 * k);
                // A = a_src broadcast over M, B = x tile -> every D row = matvec result
                c = __builtin_amdgcn_wmma_f32_16x16x4_f32(
                        false, asv[k], false, xv, (short)0, c, false, false);
            }
            float ev = tgt + c[0];              // D[0] lane l == result for column l&15
            ev = ev > 0.0f ? ev : 0.01f * ev;   // leaky_relu(0.01)
            const int mk = adj_mask[n0 + lo];
            if (lane < 16) {
                e_out[(size_t)b * N + n0 + lane] = ev;
                if (mk > 0) vals[i] = ev;       // lanes >=16 are duplicates: excluded
            }
        }
    }

    // Per-lane local max/sum
    float m = NEGBIG, s = 0.0f;
    #pragma unroll
    for (int i = 0; i < TPW; ++i) m = fmaxf(m, vals[i]);
    #pragma unroll
    for (int i = 0; i < TPW; ++i) s += __expf(vals[i] - m);
    if (lane >= 16) { m = NEGBIG; s = 0.0f; }   // kill duplicate half-wave

    // Wave reduction (online-softmax combine)
    #pragma unroll
    for (int off = 16; off > 0; off >>= 1) {
        float m2 = __shfl_xor(m, off, 32);
        float s2 = __shfl_xor(s, off, 32);
        online_combine(m, s, m2, s2);
    }

    // Block reduction across 8 waves
    __shared__ float2 red[WAVES];
    if (lane == 0) red[wave] = make_float2(m, s);
    __syncthreads();
    if (wave == 0) {
        float mm = (lane < WAVES) ? red[lane].x : NEGBIG;
        float ss = (lane < WAVES) ? red[lane].y : 0.0f;
        #pragma unroll
        for (int off = 4; off > 0; off >>= 1) {
            float m2 = __shfl_xor(mm, off, 32);
            float s2 = __shfl_xor(ss, off, 32);
            online_combine(mm, ss, m2, s2);
        }
        if (lane == 0) partials[(size_t)b * nchunks + chunk] = make_float2(mm, ss);
    }
}

// Kernel 2: reduce per-chunk partials -> finals[b] = (max, sum). One wave per b.
__global__ void gat_reduce_kernel(const float2* __restrict__ partials,
                                  float2*       __restrict__ finals,
                                  int nchunks)
{
    const int b    = blockIdx.x;
    const int lane = threadIdx.x;
    float m = NEGBIG, s = 0.0f;
    for (int c = lane; c < nchunks; c += 32) {
        float2 p = partials[(size_t)b * nchunks + c];
        online_combine(m, s, p.x, p.y);
    }
    #pragma unroll
    for (int off = 16; off > 0; off >>= 1) {
        float m2 = __shfl_xor(m, off, 32);
        float s2 = __shfl_xor(s, off, 32);
        online_combine(m, s, m2, s2);
    }
    if (lane == 0) finals[b] = make_float2(m, s);
}

// Kernel 3: out[b,n] = mask[n] * exp(e[b,n] - M_b) / S_b  (0 where masked)
__global__ void gat_out_kernel(const float*  __restrict__ e,
                               const int*    __restrict__ adj_mask,
                               const float2* __restrict__ finals,
                               float*        __restrict__ out,
                               int N)
{
    const int b = blockIdx.y;
    const size_t i = (size_t)blockIdx.x * blockDim.x + threadIdx.x;
    if (i >= (size_t)N) return;
    const float2 f = finals[b];
    const int mk = adj_mask[i];
    float r = 0.0f;
    if (mk > 0) r = __expf(e[(size_t)b * N + i] - f.x) / f.y * (float)mk;
    out[(size_t)b * N + i] = r;
}

extern "C" void kernel_launch(void* const* d_in, const int* in_sizes, int n_in,
                              void* d_out, int out_size, void* d_ws, size_t ws_size,
                              hipStream_t stream)
{
    const float* x          = (const float*)d_in[0];
    const float* a          = (const float*)d_in[1];
    const int*   node_index = (const int*)d_in[2];
    const int*   adj_mask   = (const int*)d_in[3];
    float*       out        = (float*)d_out;

    const int N = in_sizes[3];                       // 50000
    const int D = in_sizes[1] / 2;                   // 64 (layout hardcoded for 64)
    const int B = (int)((long long)in_sizes[0] / ((long long)N * D));
    const int nchunks = (N + ROWS_PER_BLOCK - 1) / ROWS_PER_BLOCK;

    float*  ws_e     = (float*)d_ws;                          // B*N floats
    float2* partials = (float2*)(ws_e + (size_t)B * N);       // B*nchunks
    float2* finals   = partials + (size_t)B * nchunks;        // B

    dim3 g1(nchunks, B);
    gat_scores_kernel<<<g1, dim3(WAVES * 32), 0, stream>>>(
        x, a, node_index, adj_mask, ws_e, partials, N, nchunks);

    gat_reduce_kernel<<<dim3(B), dim3(32), 0, stream>>>(partials, finals, nchunks);

    dim3 g3((N + 255) / 256, B);
    gat_out_kernel<<<g3, dim3(256), 0, stream>>>(ws_e, adj_mask, finals, out, N);
}